// BSpline_56049323212965
// MI455X (gfx1250) — compile-verified
//
#include <hip/hip_runtime.h>
#include <stdint.h>

// ---------------------------------------------------------------------------
// B-spline scatter, MI455X (gfx1250).
//
// Roofline: 1 GiB of output stores dominates (~46 us at 23.3 TB/s HBM);
// compute is ~3 MFLOP. Strategy:
//   Kernel 1: bulk zero-fill of d_out via CDNA5 async store-from-LDS
//             (global_store_async_from_lds_b128, ASYNCcnt-tracked) with
//             th:TH_STORE_NT (output is 5x the 192MB L2 -> stream it).
//   Kernel 2: the reference einsum values[s,k] = sum_p powers[s,k,p]*B[k,p]
//             done on the matrix pipe with V_WMMA_F32_16X16X4_F32 (f32 in,
//             f32 acc -> bit-compatible precision class with the reference).
//             Basis rebased onto powers of frac so one WMMA = 16 samples:
//               values[s,k] = sum_p C[k,p] * frac_s^p,
//               C[k,p] = sum_{m>=p} B[k,m]*binom(m,p)*(q-k)^(m-p).
//             D columns 0..3 hold the 4 basis values; LDS bounce, then the
//             4-element scatter (distinct columns per row -> plain stores).
// ---------------------------------------------------------------------------

#define QORD 3

typedef float v2f __attribute__((ext_vector_type(2)));
typedef float v8f __attribute__((ext_vector_type(8)));

// ---- Kernel 1: zero-fill using async store-from-LDS (CDNA5 path) ----------
__global__ void __launch_bounds__(256)
bspline_zero_fill_async(float* __restrict__ out, long long n_floats)
{
    __shared__ __align__(16) float zbuf[128];   // 512 B of zeros
    if (threadIdx.x < 128) zbuf[threadIdx.x] = 0.0f;
    __syncthreads();

    const unsigned lane = threadIdx.x & 31u;
    // addrspace(3)->generic keeps the LDS byte offset in the low 32 bits.
    const unsigned lds_off =
        (unsigned)(unsigned long long)(const void*)(&zbuf[lane * 4u]);

    const long long tid      = (long long)blockIdx.x * blockDim.x + threadIdx.x;
    const long long nthreads = (long long)gridDim.x * blockDim.x;

    // 64 B per thread per iteration (4 x b128); wave covers contiguous 2 KiB.
    const long long niter = n_floats / 16;
    for (long long i = tid; i < niter; i += nthreads) {
        float* p = out + i * 16;
        asm volatile(
            "global_store_async_from_lds_b128 %0, %1, off offset:0  th:TH_STORE_NT\n\t"
            "global_store_async_from_lds_b128 %0, %1, off offset:16 th:TH_STORE_NT\n\t"
            "global_store_async_from_lds_b128 %0, %1, off offset:32 th:TH_STORE_NT\n\t"
            "global_store_async_from_lds_b128 %0, %1, off offset:48 th:TH_STORE_NT"
            :: "v"(p), "v"(lds_off) : "memory");
    }

    const long long tail = niter * 16;
    for (long long j = tail + tid; j < n_floats; j += nthreads) out[j] = 0.0f;

    asm volatile("s_wait_asynccnt 0" ::: "memory");
}

// ---- Kernel 2: WMMA einsum + scatter (16 samples per wave-WMMA) -----------
__global__ void __launch_bounds__(256)
bspline_wmma_scatter(const float* __restrict__ xs,
                     const float* __restrict__ Bc,   // [4][4] ascending powers
                     const int*   __restrict__ n_ptr,
                     float*       __restrict__ out,
                     int nsamp)
{
    __shared__ float vals[8][16][4];                 // 8 waves x 16 samples x 4
    const int lane  = threadIdx.x & 31;
    const int wv    = threadIdx.x >> 5;
    const int s_loc = lane & 15;

    const int base = (blockIdx.x * 8 + wv) * 16;     // 16 samples per wave
    const int nk   = *n_ptr;                         // uniform -> scalar load

    // ---- per-lane sample (sample s owned by lanes s and s+16); clamp the
    //      load index so EXEC stays all-ones through the WMMA on tail batches.
    const int   sidx   = base + s_loc;
    const int   sload  = (sidx < nsamp) ? sidx : (nsamp - 1);
    const float x      = xs[sload];
    const float fi     = floorf(x);                  // H=1, T0=0
    const float frac   = x - fi;

    // ---- A matrix 16x4 f32 (ISA layout: lanes<16 -> K=0,1; lanes>=16 -> K=2,3)
    v2f a;
    if (lane < 16) { a.x = 1.0f; a.y = frac; }
    else           { const float f2 = frac * frac; a.x = f2; a.y = f2 * frac; }

    // ---- B matrix 4x16 f32: column N carries shifted coeff row k=N%4.
    //      C[k][p] via binomial shift by t = (q-k)*H:
    //        C0 = B0 + B1 t + B2 t^2 + B3 t^3
    //        C1 = B1 + 2 B2 t + 3 B3 t^2
    //        C2 = B2 + 3 B3 t
    //        C3 = B3
    const int   kcol = s_loc & 3;
    const float t    = (float)(QORD - kcol);
    const float b0 = Bc[kcol * 4 + 0], b1 = Bc[kcol * 4 + 1];
    const float b2 = Bc[kcol * 4 + 2], b3 = Bc[kcol * 4 + 3];
    const float c0 = fmaf(fmaf(fmaf(b3, t, b2), t, b1), t, b0);
    const float c1 = fmaf(fmaf(3.0f * b3, t, 2.0f * b2), t, b1);
    const float c2 = fmaf(3.0f * b3, t, b2);
    const float c3 = b3;
    v2f b;
    if (lane < 16) { b.x = c0; b.y = c1; }           // K rows 0,1
    else           { b.x = c2; b.y = c3; }           // K rows 2,3

    // ---- D = A x B + 0 on the matrix pipe (f32 accumulate)
    v8f cacc = {};
    v8f d = __builtin_amdgcn_wmma_f32_16x16x4_f32(
        /*neg_a=*/false, a, /*neg_b=*/false, b,
        /*c_mod=*/(short)0, cacc, /*reuse_a=*/false, /*reuse_b=*/false);

    // ---- D layout: VGPR r -> M=r (lanes 0-15, N=lane) and M=r+8
    //      (lanes 16-31, N=lane-16). Columns 0..3 give k=0..3.
    if (lane < 4) {
#pragma unroll
        for (int r = 0; r < 8; ++r) vals[wv][r][lane] = d[r];
    } else if (lane >= 16 && lane < 20) {
#pragma unroll
        for (int r = 0; r < 8; ++r) vals[wv][8 + r][lane - 16] = d[r];
    }
    __syncthreads();

    // ---- scatter: 16 lanes per wave, one sample each, 4 distinct columns
    if (lane < 16 && sidx < nsamp) {
        const int first_i = (int)fi;
        float* row = out + (long long)sidx * (long long)nk;
#pragma unroll
        for (int k = 0; k <= QORD; ++k) {
            const int col = first_i + k;
            if (col >= 0 && col < nk) row[col] = vals[wv][s_loc][k];
        }
    }
}

// ---------------------------------------------------------------------------
extern "C" void kernel_launch(void* const* d_in, const int* in_sizes, int n_in,
                              void* d_out, int out_size, void* d_ws, size_t ws_size,
                              hipStream_t stream)
{
    // setup_inputs() order: xs [N] f32, B [16] f32, n [1] i32, q [1] i32
    const float* xs  = (const float*)d_in[0];
    const float* Bc  = (const float*)d_in[1];
    const int*   n_p = (const int*)d_in[2];
    float* out = (float*)d_out;

    const int nsamp = in_sizes[0];
    const long long n_floats = (long long)out_size;

    // 1 GiB fill: 2048 blocks x 256 threads (8 wave32/block).
    bspline_zero_fill_async<<<2048, 256, 0, stream>>>(out, n_floats);

    // 16 samples per wave, 8 waves per block -> 128 samples per block.
    const int waves  = (nsamp + 15) / 16;
    const int blocks = (waves + 7) / 8;
    bspline_wmma_scatter<<<blocks, 256, 0, stream>>>(xs, Bc, n_p, out, nsamp);
}